// GCNLayer_61538291417730
// MI455X (gfx1250) — compile-verified
//
#include <hip/hip_runtime.h>
#include <stdint.h>

// ---------------------------------------------------------------------------
// GCN layer for MI455X (gfx1250, wave32):
//   hw  = (h @ W) * norm              -> d_ws   (f32 WMMA 16x16x4, async-LDS A)
//   agg = scatter_add(hw[src] -> dst) -> d_out  (global_atomic_add_f32, saddr)
//   out = relu(agg*norm + bias + h @ res_w.T + res_b)   (fused WMMA finalize)
// ---------------------------------------------------------------------------

typedef __attribute__((ext_vector_type(2))) float v2f;
typedef __attribute__((ext_vector_type(8))) float v8f;

#define NNODES 100000
#define NEDGES 1600000
#define FDIM   256
#define LSTRIDE 260   // LDS row stride in floats: 1040 B/row (16B multiple).
                      // 260 % 64 == 4 -> 16 rows hit 16 distinct 4-bank groups;
                      // with the lane-hi +2 offset all 64 banks are touched
                      // exactly once per ds_load_b64 wavefront (conflict-free).

// ---------------------------------------------------------------- zero d_out
__global__ __launch_bounds__(256) void k_zero(float* __restrict__ out, int n4) {
    int i = blockIdx.x * blockDim.x + threadIdx.x;
    if (i < n4) {
        float4 z = make_float4(0.f, 0.f, 0.f, 0.f);
        ((float4*)out)[i] = z;
    }
}

// Async stage of the 16x256 f32 A panel into LDS via ASYNCcnt-tracked copies.
// 8 waves x (2 rows x 2 half-rows) x 32 lanes x 16B = 16 KB.
__device__ __forceinline__ void stage_A_async(float* lA, const float* hbase,
                                              int wave, int lane) {
    uint32_t lbase = (uint32_t)(uintptr_t)lA;   // LDS aperture: offset in [31:0]
    #pragma unroll
    for (int j = 0; j < 4; ++j) {
        int row   = wave * 2 + (j >> 1);
        int chunk = (j & 1) * 32 + lane;        // 16B chunk index within row
        uint32_t loff = lbase + (uint32_t)(row * (LSTRIDE * 4) + chunk * 16);
        uint64_t gaddr = (uint64_t)(uintptr_t)
            ((const char*)hbase + (size_t)row * (FDIM * 4) + chunk * 16);
        asm volatile("global_load_async_to_lds_b128 %0, %1, off"
                     :: "v"(loff), "v"(gaddr) : "memory");
    }
}

// ------------------------------------------------- hw = (h @ W) * norm  (k1)
// Block: 256 threads = 8 waves, one 16-row node tile (M) per block.
// Wave w computes N-columns [w*32, w*32+32) as two 16x16 WMMA accumulators.
__global__ __launch_bounds__(256) void k_gemm_norm(
    const float* __restrict__ h, const float* __restrict__ norm,
    const float* __restrict__ weight, float* __restrict__ hw) {

    __shared__ __align__(16) float lA[16 * LSTRIDE];
    __shared__ float lN[16];

    const int m0   = blockIdx.x * 16;
    const int tid  = threadIdx.x;
    const int wave = tid >> 5;
    const int lane = tid & 31;

    stage_A_async(lA, h + (size_t)m0 * FDIM, wave, lane);
    if (tid < 16) lN[tid] = norm[m0 + tid];
    asm volatile("s_wait_asynccnt 0x0" ::: "memory");
    __syncthreads();

    const int row = lane & 15;   // A-matrix M index per ISA 16x4 f32 layout
    const int hi  = lane >> 4;   // selects K pair {0,1} vs {2,3}
    const int col = lane & 15;   // B/C/D N index within tile
    const int n0  = wave * 32;

    v8f c0 = {0.f,0.f,0.f,0.f,0.f,0.f,0.f,0.f};
    v8f c1 = {0.f,0.f,0.f,0.f,0.f,0.f,0.f,0.f};

    const float* lArow = &lA[row * LSTRIDE + hi * 2];

    #pragma unroll 4
    for (int k = 0; k < FDIM; k += 4) {
        // A 16x4 tile: VGPR0=K(k+hi*2), VGPR1=K(k+hi*2+1)  (ds_load_b64)
        v2f a;
        a.x = lArow[k];
        a.y = lArow[k + 1];

        const int kr = k + hi * 2;
        // B 4x16 tiles from row-major weight[K][N] (strided b32 loads, L2-hot)
        v2f b0, b1;
        b0.x = weight[(size_t)kr * FDIM + n0 + col];
        b0.y = weight[(size_t)(kr + 1) * FDIM + n0 + col];
        b1.x = weight[(size_t)kr * FDIM + n0 + 16 + col];
        b1.y = weight[(size_t)(kr + 1) * FDIM + n0 + 16 + col];

        c0 = __builtin_amdgcn_wmma_f32_16x16x4_f32(false, a, false, b0,
                                                   (short)0, c0, false, false);
        c1 = __builtin_amdgcn_wmma_f32_16x16x4_f32(false, a, false, b1,
                                                   (short)0, c1, false, false);
    }

    // D layout: VGPR r -> M = r + hi*8, N = col. Scale by norm and store.
    #pragma unroll
    for (int r = 0; r < 8; ++r) {
        int lrow = r + hi * 8;
        float nv = lN[lrow];
        size_t base = (size_t)(m0 + lrow) * FDIM + n0 + col;
        hw[base]      = c0[r] * nv;
        hw[base + 16] = c1[r] * nv;
    }
}

// --------------------------------------- edge scatter-add (wave per edge, k2)
__global__ __launch_bounds__(256) void k_scatter(
    const float* __restrict__ hw, const int* __restrict__ src,
    const int* __restrict__ dst, float* __restrict__ agg, int nedges) {

    int e    = (blockIdx.x * 256 + threadIdx.x) >> 5;
    int lane = threadIdx.x & 31;
    if (e >= nedges) return;

    int s = src[e];
    int d = dst[e];
    const float* srow = hw + (size_t)s * FDIM;
    uint32_t dbase = (uint32_t)d * (FDIM * 4);      // byte offset, < 2^31
    uint64_t aggbase = (uint64_t)(uintptr_t)agg;    // uniform SGPR-pair base

    #pragma unroll
    for (int i = 0; i < FDIM / 32; ++i) {
        int f = lane + i * 32;                      // coalesced 128B / iter
        float v = srow[f];
        uint32_t off = dbase + (uint32_t)f * 4;
        // Non-returning float atomic add, SADDR form (STOREcnt-tracked).
        asm volatile("global_atomic_add_f32 %0, %1, %2"
                     :: "v"(off), "v"(v), "s"(aggbase)
                     : "memory");
    }
}

// ---- out = relu(agg*norm + bias + h @ res_w.T + res_b)   (fused GEMM, k3)
__global__ __launch_bounds__(256) void k_finalize(
    const float* __restrict__ h, const float* __restrict__ norm,
    const float* __restrict__ resw, const float* __restrict__ bias,
    const float* __restrict__ resb, float* __restrict__ out) {

    __shared__ __align__(16) float lA[16 * LSTRIDE];
    __shared__ float lN[16];

    const int m0   = blockIdx.x * 16;
    const int tid  = threadIdx.x;
    const int wave = tid >> 5;
    const int lane = tid & 31;

    stage_A_async(lA, h + (size_t)m0 * FDIM, wave, lane);
    if (tid < 16) lN[tid] = norm[m0 + tid];
    asm volatile("s_wait_asynccnt 0x0" ::: "memory");
    __syncthreads();

    const int row = lane & 15;
    const int hi  = lane >> 4;
    const int col = lane & 15;
    const int n0  = wave * 32;

    v8f c0 = {0.f,0.f,0.f,0.f,0.f,0.f,0.f,0.f};
    v8f c1 = {0.f,0.f,0.f,0.f,0.f,0.f,0.f,0.f};

    const float* lArow = &lA[row * LSTRIDE + hi * 2];

    #pragma unroll 4
    for (int k = 0; k < FDIM; k += 4) {
        v2f a;
        a.x = lArow[k];
        a.y = lArow[k + 1];

        const int kr = k + hi * 2;
        // B[k][n] = res_w[n][k]: row n of res_w is contiguous in K -> b64 loads
        v2f b0 = *(const v2f*)(resw + (size_t)(n0 + col) * FDIM + kr);
        v2f b1 = *(const v2f*)(resw + (size_t)(n0 + 16 + col) * FDIM + kr);

        c0 = __builtin_amdgcn_wmma_f32_16x16x4_f32(false, a, false, b0,
                                                   (short)0, c0, false, false);
        c1 = __builtin_amdgcn_wmma_f32_16x16x4_f32(false, a, false, b1,
                                                   (short)0, c1, false, false);
    }

    float bias0 = bias[n0 + col]      + resb[n0 + col];
    float bias1 = bias[n0 + 16 + col] + resb[n0 + 16 + col];

    #pragma unroll
    for (int r = 0; r < 8; ++r) {
        int lrow = r + hi * 8;
        float nv = lN[lrow];
        size_t base = (size_t)(m0 + lrow) * FDIM + n0 + col;

        float v0 = out[base]      * nv + bias0 + c0[r];
        float v1 = out[base + 16] * nv + bias1 + c1[r];
        out[base]      = fmaxf(v0, 0.f);
        out[base + 16] = fmaxf(v1, 0.f);
    }
}

// ---------------------------------------------------------------------------
extern "C" void kernel_launch(void* const* d_in, const int* in_sizes, int n_in,
                              void* d_out, int out_size, void* d_ws, size_t ws_size,
                              hipStream_t stream) {
    const float* h      = (const float*)d_in[0];
    const float* norm   = (const float*)d_in[1];
    const int*   src    = (const int*)  d_in[2];
    const int*   dst    = (const int*)  d_in[3];
    const float* weight = (const float*)d_in[4];
    const float* bias   = (const float*)d_in[5];
    const float* resw   = (const float*)d_in[6];
    const float* resb   = (const float*)d_in[7];

    float* out = (float*)d_out;
    float* hw  = (float*)d_ws;          // needs NNODES*FDIM*4 = 102.4 MB

    const int n4 = NNODES * FDIM / 4;   // 6,400,000 float4
    k_zero<<<(n4 + 255) / 256, 256, 0, stream>>>(out, n4);

    k_gemm_norm<<<NNODES / 16, 256, 0, stream>>>(h, norm, weight, hw);

    k_scatter<<<NEDGES / 8, 256, 0, stream>>>(hw, src, dst, out, NEDGES);

    k_finalize<<<NNODES / 16, 256, 0, stream>>>(h, norm, resw, bias, resb, out);
}